// SuperPixelMeanEmbed_61813169324342
// MI455X (gfx1250) — compile-verified
//
#include <hip/hip_runtime.h>
#include <hip/hip_bf16.h>

typedef float v2f __attribute__((ext_vector_type(2)));
typedef float v8f __attribute__((ext_vector_type(8)));

#define D_DIM   12288      // C*H*H = 3*64*64
#define B_DIM   32         // batch
#define HW_DIM  4096       // H*H
#define C_DIM   3
#define NSEG    196
#define KC      64         // K chunk staged in LDS per round
#define N_TILE  128        // W rows per workgroup (8 waves x 16 rows)
#define KSPLIT  4          // split-K factor (global fp32 atomics accumulate)
#define WPITCH  68         // LDS pitch for W tile (mult of 4 -> aligned b128; bank-friendly)
#define XP2     33         // LDS pitch (in float2 units) for interleaved X^T chunk

// Async global->LDS copy of 16 bytes per lane (CDNA5, tracked by ASYNCcnt).
// LDS address = low 32 bits of the flat shared pointer (flat->LDS truncation, ISA 10.2).
// th:TH_LOAD_NT : W is streamed once (604 MB) -- keep it out of the 192 MB L2.
__device__ __forceinline__ void async_copy_b128_nt(const float* gsrc, float* ldst) {
  unsigned int l = (unsigned int)(unsigned long long)ldst;
  asm volatile("global_load_async_to_lds_b128 %0, %1, off th:TH_LOAD_NT"
               :: "v"(l), "v"(gsrc) : "memory");
}
__device__ __forceinline__ void wait_asynccnt0() {
  asm volatile("s_wait_asynccnt 0x0" ::: "memory");
}

// ---------------------------------------------------------------------------
// GEMM: Y[b][n] += sum_k X[b][k] * W[n][k]   (bias added later in pooling)
// WMMA orientation: D[n(16), b(16)] = A(W rows 16xK) x B(X^T Kx16) + C
// A layout (f32 16x16x4): lane<16: row=lane, K={k,k+1}; lane>=16: row=lane-16, K={k+2,k+3}
// B layout assumed symmetric: V0/V1 = B[K]/B[K+1] lanes<16, B[K+2]/B[K+3] lanes>=16
// C/D layout: VGPR i, lane L: M = i + 8*(L/16), N = L%16
// ---------------------------------------------------------------------------
__global__ __launch_bounds__(256)
void SuperPixel_gemm_wmma(const float* __restrict__ X,
                          const float* __restrict__ W,
                          float* __restrict__ Y) {
  // Xs FIRST: keeps the X buffers in LDS [0, 16896) so every B-operand
  // ds_load address fits the 16-bit DS immediate-offset field.
  __shared__ __align__(16) float Xs[2][(KC / 2) * XP2 * 2];
  __shared__ __align__(16) float Wt[2][N_TILE * WPITCH];

  const int tid  = threadIdx.x;
  const int lane = tid & 31;
  const int wv   = tid >> 5;            // 0..7
  const int l15  = lane & 15;
  const int hb   = lane >> 4;           // 0/1: K+2 half for A/B, M+8 half for D
  const int h2   = hb << 1;

  const int n0   = blockIdx.x * N_TILE;
  const int kseg = D_DIM / KSPLIT;      // 3072 (48 chunks of KC=64, even)
  const int kbeg = blockIdx.y * kseg;

  v8f acc0 = {0.f, 0.f, 0.f, 0.f, 0.f, 0.f, 0.f, 0.f};   // batches 0..15
  v8f acc1 = {0.f, 0.f, 0.f, 0.f, 0.f, 0.f, 0.f, 0.f};   // batches 16..31

  // async-stage one K-chunk of the W tile (128 rows x 64 cols, b128 per lane)
  auto stage_W = [&](int buf, int k0) {
    for (int e4 = tid; e4 < (N_TILE * KC) / 4; e4 += 256) {
      const int idx = e4 << 2;
      const int row = idx / KC;
      const int col = idx & (KC - 1);
      async_copy_b128_nt(&W[(size_t)(n0 + row) * D_DIM + k0 + col],
                         &Wt[buf][row * WPITCH + col]);
    }
  };
  // stage X^T chunk, K-pair interleaved: Xs[(k/2)*XP2 + b] = {x[k], x[k+1]}
  auto stage_X = [&](int buf, int k0) {
    for (int e = tid; e < B_DIM * KC; e += 256) {
      const int bb = e >> 6;            // /KC
      const int kk = e & (KC - 1);
      Xs[buf][(((kk >> 1) * XP2) + bb) * 2 + (kk & 1)] =
          X[(size_t)bb * D_DIM + k0 + kk];
    }
  };
  // consume one staged chunk: 16 K-steps x 2 WMMA (buf is a compile-time 0/1)
  auto compute = [&](int buf) {
    const float* wrow = &Wt[buf][(wv * 16 + l15) * WPITCH];
    const float* xb   = &Xs[buf][0];
#pragma unroll
    for (int kk = 0; kk < KC; kk += 4) {
      const int ka = kk + h2;                       // even
      const v2f a  = *(const v2f*)&wrow[ka];        // (W[row][ka], W[row][ka+1])
      const int kp = ka >> 1;
      const v2f b0 = *(const v2f*)&xb[(kp * XP2 + l15) * 2];        // b 0..15
      const v2f b1 = *(const v2f*)&xb[(kp * XP2 + 16 + l15) * 2];   // b 16..31
      acc0 = __builtin_amdgcn_wmma_f32_16x16x4_f32(false, a, false, b0,
                                                   (short)0, acc0, false, false);
      acc1 = __builtin_amdgcn_wmma_f32_16x16x4_f32(false, a, false, b1,
                                                   (short)0, acc1, false, false);
    }
  };

  stage_W(0, kbeg);
  stage_X(0, kbeg);

  // ping-pong with compile-time buffer indices (kseg/KC = 48 chunks, even)
  for (int kc = 0; kc < kseg; kc += 2 * KC) {
    wait_asynccnt0();                    // buffer 0 async loads landed
    __syncthreads();
    if (kc + KC < kseg) { stage_W(1, kbeg + kc + KC); stage_X(1, kbeg + kc + KC); }
    compute(0);

    wait_asynccnt0();                    // buffer 1 async loads landed
    __syncthreads();
    if (kc + 2 * KC < kseg) { stage_W(0, kbeg + kc + 2 * KC); stage_X(0, kbeg + kc + 2 * KC); }
    compute(1);
  }

  // split-K accumulate into Y with fp32 global atomics
  const int rbase = n0 + wv * 16 + 8 * hb;
#pragma unroll
  for (int i = 0; i < 8; ++i) {
    const int rg = rbase + i;
    __hip_atomic_fetch_add(&Y[(size_t)l15 * D_DIM + rg], acc0[i],
                           __ATOMIC_RELAXED, __HIP_MEMORY_SCOPE_AGENT);
    __hip_atomic_fetch_add(&Y[(size_t)(16 + l15) * D_DIM + rg], acc1[i],
                           __ATOMIC_RELAXED, __HIP_MEMORY_SCOPE_AGENT);
  }
}

// ---------------------------------------------------------------------------
// Per-batch superpixel mean: out[b][s][c] = mean over pixels p with seg==s of
//   (Y[b][c*HW + p] + bias[c*HW + p])
// ---------------------------------------------------------------------------
__global__ __launch_bounds__(256)
void SuperPixel_segmean(const float* __restrict__ Y,
                        const float* __restrict__ bias,
                        const int* __restrict__ seg,
                        float* __restrict__ out) {
  __shared__ float sums[NSEG * C_DIM];
  __shared__ float cnt[NSEG];
  const int b   = blockIdx.x;
  const int tid = threadIdx.x;

  for (int i = tid; i < NSEG * C_DIM; i += 256) sums[i] = 0.f;
  for (int i = tid; i < NSEG; i += 256) cnt[i] = 0.f;
  __syncthreads();

  for (int p = tid; p < HW_DIM; p += 256) {
    const int s = seg[b * HW_DIM + p];
    atomicAdd(&cnt[s], 1.0f);
#pragma unroll
    for (int c = 0; c < C_DIM; ++c) {
      const float v = Y[(size_t)b * D_DIM + c * HW_DIM + p] +
                      bias[c * HW_DIM + p];
      atomicAdd(&sums[s * C_DIM + c], v);
    }
  }
  __syncthreads();

  for (int s = tid; s < NSEG; s += 256) {
    const float inv = 1.0f / fmaxf(cnt[s], 1.0f);
#pragma unroll
    for (int c = 0; c < C_DIM; ++c)
      out[((size_t)b * NSEG + s) * C_DIM + c] = sums[s * C_DIM + c] * inv;
  }
}

extern "C" void kernel_launch(void* const* d_in, const int* in_sizes, int n_in,
                              void* d_out, int out_size, void* d_ws, size_t ws_size,
                              hipStream_t stream) {
  const float* X    = (const float*)d_in[0];   // [32,3,64,64] == [32,12288]
  const float* W    = (const float*)d_in[1];   // [12288,12288]
  const float* bias = (const float*)d_in[2];   // [12288]
  const int*   seg  = (const int*)d_in[3];     // [32,64,64]
  float*       out  = (float*)d_out;           // [32,196,3]
  float*       Y    = (float*)d_ws;            // [32,12288] fp32 accumulator

  (void)in_sizes; (void)n_in; (void)out_size; (void)ws_size;

  // zero the split-K accumulator (graph-capturable async memset)
  hipMemsetAsync(Y, 0, (size_t)B_DIM * D_DIM * sizeof(float), stream);

  dim3 ggrid(D_DIM / N_TILE, KSPLIT);          // 96 x 4
  SuperPixel_gemm_wmma<<<ggrid, 256, 0, stream>>>(X, W, Y);

  SuperPixel_segmean<<<B_DIM, 256, 0, stream>>>(Y, bias, seg, out);
}